// EdgeEnhancedGATConv_84559316124081
// MI455X (gfx1250) — compile-verified
//
#include <hip/hip_runtime.h>
#include <hip/hip_bf16.h>
#include <math.h>

#define Nn 100000
#define Ee 1600000
#define Hh 4
#define Dd 32
#define HD 128
#define NEG_SLOPE 0.2f

typedef __attribute__((ext_vector_type(2))) float v2f;
typedef __attribute__((ext_vector_type(8))) float v8f;

// Monotone map float -> uint so that unsigned atomicMax == float max.
__device__ __forceinline__ unsigned fkey(float f) {
  unsigned u = __float_as_uint(f);
  return (u & 0x80000000u) ? ~u : (u | 0x80000000u);
}
__device__ __forceinline__ float fkeyinv(unsigned k) {
  unsigned u = (k & 0x80000000u) ? (k & 0x7fffffffu) : ~k;
  return __uint_as_float(u);
}
#define KEY_NEG_INF 0x007FFFFFu  // fkey(-inf)

__device__ __forceinline__ float lrelu(float x) { return x > 0.f ? x : NEG_SLOPE * x; }

__device__ __forceinline__ void edge_sd(long long i, const int* __restrict__ src,
                                        const int* __restrict__ dst, int& s, int& d) {
  if (i < (long long)Ee) { s = src[i]; d = dst[i]; }
  else { s = d = (int)(i - (long long)Ee); }  // appended self-loops
}

// ---------------------------------------------------------------- init
__global__ void __launch_bounds__(256)
k_init(float* __restrict__ out, const float* __restrict__ bias,
       unsigned* __restrict__ emax, float* __restrict__ denom) {
  long long tid = (long long)blockIdx.x * 256 + threadIdx.x;
  if (tid < (long long)Nn * HD) out[tid] = bias[tid & (HD - 1)];
  if (tid < (long long)Nn * Hh) { emax[tid] = KEY_NEG_INF; denom[tid] = 0.f; }
}

// ---------------------------------------------------------------- GEMM (WMMA f32)
// feat[N,128] = node_feats[N,128] @ W_fc[128,128], f32 via V_WMMA_F32_16X16X4_F32.
// Block = 8 waves; block owns a 16-row tile of A (staged in LDS, padded stride),
// wave w owns output columns [16w, 16w+16). K accumulated in steps of 4.
__global__ void __launch_bounds__(256)
k_gemm(const float* __restrict__ A, const float* __restrict__ W, float* __restrict__ C) {
  __shared__ float At[16 * 132];  // 132-stride padding -> conflict-free column reads
  const int m0 = blockIdx.x * 16;

  // coalesced float4 stage of the 16x128 A tile
  for (int t = threadIdx.x; t < 16 * 32; t += 256) {
    const int row = t >> 5;
    const int c4  = (t & 31) << 2;
    const float4 v = *(const float4*)(A + (long long)(m0 + row) * HD + c4);
    float* p = At + row * 132 + c4;
    p[0] = v.x; p[1] = v.y; p[2] = v.z; p[3] = v.w;
  }
  __syncthreads();

  const int lane = threadIdx.x & 31;
  const int n0   = (threadIdx.x >> 5) << 4;   // wave's column tile
  const int lr   = lane & 15;
  const int koff = (lane >> 4) << 1;          // lanes 0-15: K0/K1, lanes 16-31: K2/K3

  v8f acc = {};
  const float* arow = At + lr * 132 + koff;                 // A: row lr, K slice
  const float* bcol = W + (long long)koff * HD + n0 + lr;   // B: col n0+lr, K slice
#pragma unroll
  for (int k = 0; k < 128; k += 4) {
    v2f a = { arow[k], arow[k + 1] };
    v2f b = { bcol[(long long)k * HD], bcol[(long long)(k + 1) * HD] };
    acc = __builtin_amdgcn_wmma_f32_16x16x4_f32(false, a, false, b, (short)0, acc,
                                                false, false);
  }

  // D layout: VGPR r -> row r (lanes 0-15) / row r+8 (lanes 16-31), col = lane&15
  const int col   = n0 + lr;
  const int rbase = m0 + ((lane >> 4) << 3);
#pragma unroll
  for (int r = 0; r < 8; ++r)
    C[(long long)(rbase + r) * HD + col] = acc[r];
}

// ---------------------------------------------------------------- el / er
__global__ void __launch_bounds__(256)
k_attn(const float* __restrict__ feat, const float* __restrict__ al,
       const float* __restrict__ ar, float* __restrict__ el, float* __restrict__ er) {
  int tid = blockIdx.x * 256 + threadIdx.x;
  if (tid >= Nn * Hh) return;
  const int h = tid & 3;
  const float* f = feat + (long long)(tid >> 2) * HD + h * Dd;
  const float* l = al + h * Dd;
  const float* r = ar + h * Dd;
  float sl = 0.f, sr = 0.f;
#pragma unroll
  for (int d = 0; d < Dd; ++d) { sl = fmaf(f[d], l[d], sl); sr = fmaf(f[d], r[d], sr); }
  el[tid] = sl; er[tid] = sr;
}

// ---------------------------------------------------------------- segment max
__global__ void __launch_bounds__(256)
k_max(const int* __restrict__ src, const int* __restrict__ dst,
      const float* __restrict__ el, const float* __restrict__ er,
      unsigned* __restrict__ emax) {
  long long tid = (long long)blockIdx.x * 256 + threadIdx.x;
  if (tid >= (long long)(Ee + Nn) * Hh) return;
  const long long i = tid >> 2; const int h = (int)(tid & 3);
  int s, d; edge_sd(i, src, dst, s, d);
  const float e = lrelu(el[(long long)s * Hh + h] + er[(long long)d * Hh + h]);
  atomicMax(&emax[(long long)d * Hh + h], fkey(e));
}

// ---------------------------------------------------------------- segment sum of exp
__global__ void __launch_bounds__(256)
k_denom(const int* __restrict__ src, const int* __restrict__ dst,
        const float* __restrict__ el, const float* __restrict__ er,
        const unsigned* __restrict__ emax, float* __restrict__ denom) {
  long long tid = (long long)blockIdx.x * 256 + threadIdx.x;
  if (tid >= (long long)(Ee + Nn) * Hh) return;
  const long long i = tid >> 2; const int h = (int)(tid & 3);
  int s, d; edge_sd(i, src, dst, s, d);
  const float e = lrelu(el[(long long)s * Hh + h] + er[(long long)d * Hh + h]);
  const float ee = expf(e - fkeyinv(emax[(long long)d * Hh + h]));
  atomicAdd(&denom[(long long)d * Hh + h], ee);
}

// ---------------------------------------------------------------- weighted aggregation
// 32 threads per edge; thread j handles float4 chunk j of the 128-float message.
__global__ void __launch_bounds__(256)
k_agg(const int* __restrict__ src, const int* __restrict__ dst,
      const float* __restrict__ el, const float* __restrict__ er,
      const unsigned* __restrict__ emax, const float* __restrict__ denom,
      const float* __restrict__ feat, float* __restrict__ out) {
  long long tid = (long long)blockIdx.x * 256 + threadIdx.x;
  if (tid >= (long long)(Ee + Nn) * 32) return;
  const long long i = tid >> 5;
  const int j = (int)(tid & 31);
  const int h = j >> 3;  // 8 float4 chunks per head
  int s, d; edge_sd(i, src, dst, s, d);
  const float e = lrelu(el[(long long)s * Hh + h] + er[(long long)d * Hh + h]);
  const long long dh = (long long)d * Hh + h;
  const float alpha = expf(e - fkeyinv(emax[dh])) / denom[dh];
  const float4 f = *(const float4*)(feat + (long long)s * HD + j * 4);
  float* o = out + (long long)d * HD + j * 4;
  atomicAdd(o + 0, alpha * f.x);
  atomicAdd(o + 1, alpha * f.y);
  atomicAdd(o + 2, alpha * f.z);
  atomicAdd(o + 3, alpha * f.w);
}

extern "C" void kernel_launch(void* const* d_in, const int* in_sizes, int n_in,
                              void* d_out, int out_size, void* d_ws, size_t ws_size,
                              hipStream_t stream) {
  const float* node_feats = (const float*)d_in[0];
  // d_in[1] edge_feats, d_in[4] W_edge, d_in[5] b_edge: dead in the reference.
  const int*   src    = (const int*)d_in[2];
  const int*   dst    = (const int*)d_in[3];
  const float* W_fc   = (const float*)d_in[6];
  const float* attn_l = (const float*)d_in[7];
  const float* attn_r = (const float*)d_in[8];
  const float* bias   = (const float*)d_in[9];
  float* out = (float*)d_out;

  // workspace layout (all 16B-aligned): feat | el | er | emax | denom  (~58 MB)
  float*    feat  = (float*)d_ws;
  float*    el    = feat + (long long)Nn * HD;
  float*    er    = el + (long long)Nn * Hh;
  unsigned* emax  = (unsigned*)(er + (long long)Nn * Hh);
  float*    denom = (float*)(emax + (long long)Nn * Hh);

  const long long initT = (long long)Nn * HD;
  k_init<<<(unsigned)((initT + 255) / 256), 256, 0, stream>>>(out, bias, emax, denom);

  k_gemm<<<Nn / 16, 256, 0, stream>>>(node_feats, W_fc, feat);

  k_attn<<<(Nn * Hh + 255) / 256, 256, 0, stream>>>(feat, attn_l, attn_r, el, er);

  const long long TE = (long long)(Ee + Nn);
  k_max<<<(unsigned)((TE * Hh + 255) / 256), 256, 0, stream>>>(src, dst, el, er, emax);
  k_denom<<<(unsigned)((TE * Hh + 255) / 256), 256, 0, stream>>>(src, dst, el, er, emax, denom);
  k_agg<<<(unsigned)((TE * 32 + 255) / 256), 256, 0, stream>>>(src, dst, el, er, emax, denom,
                                                               feat, out);
}